// RNN_39934605918507
// MI455X (gfx1250) — compile-verified
//
#include <hip/hip_runtime.h>

typedef float v2f __attribute__((ext_vector_type(2)));
typedef float v8f __attribute__((ext_vector_type(8)));

#define B_  256
#define S_  512
#define I_  290
#define H_  29
#define O_  10
#define KP  292   // I_ padded to multiple of 4 (k-step granularity)
#define HP  32    // H_ padded to 32 (xp row stride, lane count)

// ---------------------------------------------------------------------------
// Kernel 1: input projection  xp[m][h] = x[m][:] . W_ih[h][:] + (b_ih+b_hh)[h]
// m = b*S + s  (M = 131072), via V_WMMA_F32_16X16X4_F32.
// 256 threads = 8 waves; wave w handles M-tile of 16 rows, both N-tiles.
// Grid covers 128 rows per block -> 1024 blocks exactly.
// ---------------------------------------------------------------------------
__global__ __launch_bounds__(256)
void k_inproj(const float* __restrict__ x, const float* __restrict__ Wih,
              const float* __restrict__ bih, const float* __restrict__ bhh,
              float* __restrict__ xp)
{
    __shared__ float wlds[HP * KP];   // 32*292*4 = 37376 B, zero padded

    const int tid = threadIdx.x;
    for (int idx = tid; idx < HP * KP; idx += 256) {
        int h = idx / KP, k = idx - h * KP;
        float v = 0.0f;
        if (h < H_ && k < I_) v = Wih[h * I_ + k];
        wlds[idx] = v;
    }
    __syncthreads();

    const int wave  = tid >> 5;
    const int lane  = tid & 31;
    const int nlane = lane & 15;
    const int khalf = lane >> 4;              // 0: K={0,1}, 1: K={2,3}
    const int m0    = blockIdx.x * 128 + wave * 16;
    const float* xrow = x + (size_t)(m0 + nlane) * I_;

    const float* bl0 = &wlds[(nlane)      * KP + 2 * khalf];   // N-tile 0
    const float* bl1 = &wlds[(nlane + 16) * KP + 2 * khalf];   // N-tile 1

    v8f acc0 = {};
    v8f acc1 = {};

    #pragma unroll 4
    for (int kk = 0; kk < 72; ++kk) {
        v2f a  = *(const v2f*)(xrow + kk * 4 + 2 * khalf);
        v2f b0 = *(const v2f*)(bl0 + kk * 4);
        v2f b1 = *(const v2f*)(bl1 + kk * 4);
        acc0 = __builtin_amdgcn_wmma_f32_16x16x4_f32(false, a, false, b0,
                                                     (short)0, acc0, false, false);
        acc1 = __builtin_amdgcn_wmma_f32_16x16x4_f32(false, a, false, b1,
                                                     (short)0, acc1, false, false);
    }
    // Tail k-step: K = 288..291; only 288,289 exist in x (lanes 0-15), rest 0.
    {
        // upper half-lanes clamp address into the row, then zero the value
        const float* atl = xrow + (khalf ? 0 : 288);
        v2f a = *(const v2f*)atl;
        if (khalf) { a.x = 0.0f; a.y = 0.0f; }        // v_cndmask, EXEC untouched
        v2f b0 = *(const v2f*)(bl0 + 288);            // LDS zero padding covers 290/291
        v2f b1 = *(const v2f*)(bl1 + 288);
        acc0 = __builtin_amdgcn_wmma_f32_16x16x4_f32(false, a, false, b0,
                                                     (short)0, acc0, false, false);
        acc1 = __builtin_amdgcn_wmma_f32_16x16x4_f32(false, a, false, b1,
                                                     (short)0, acc1, false, false);
    }

    // Bias + store.  C/D layout: lane c, reg r -> (M = r + 8*(c>>4), N = c&15).
    float bias0 = bih[nlane] + bhh[nlane];            // cols 0..15 always valid
    int   c1    = nlane + 16;
    float bias1 = (c1 < H_) ? (bih[c1] + bhh[c1]) : 0.0f;

    const int mbase = m0 + 8 * khalf;
    #pragma unroll
    for (int r = 0; r < 8; ++r) {
        size_t row = (size_t)(mbase + r) * HP;
        xp[row + nlane]      = acc0[r] + bias0;
        xp[row + nlane + 16] = acc1[r] + bias1;       // cols 29..31 = padding
    }
}

// ---------------------------------------------------------------------------
// tanh via v_exp_f32 + v_rcp_f32, saturating form (no NaN at +-inf)
// ---------------------------------------------------------------------------
__device__ __forceinline__ float tanh_fast(float a)
{
    // 2*log2(e)
    float e = __builtin_amdgcn_exp2f(a * 2.8853900817779268f);
    return 1.0f - 2.0f * __builtin_amdgcn_rcpf(e + 1.0f);
}

// ---------------------------------------------------------------------------
// Kernel 2: recurrence + final FC.  One wave per batch row; lane = hidden idx.
// h_new[h] = tanh(xp[b][t][h] + sum_k W_hh[h][k] * h_old[k])
// Broadcast of h_old via v_readlane; xp prefetched 8 steps ahead.
// ---------------------------------------------------------------------------
__global__ __launch_bounds__(32)
void k_scan(const float* __restrict__ xp, const float* __restrict__ Whh,
            const float* __restrict__ Wfc, const float* __restrict__ bfc,
            float* __restrict__ out)
{
    const int b    = blockIdx.x;
    const int lane = threadIdx.x;                 // hidden index (0..31)

    // Per-lane row of W_hh, zero padded for lanes/cols >= 29
    float w[H_];
    #pragma unroll
    for (int k = 0; k < H_; ++k)
        w[k] = (lane < H_) ? Whh[lane * H_ + k] : 0.0f;

    const float* xrow = xp + (size_t)b * S_ * HP + lane;

    float h = 0.0f;
    constexpr int U = 8;
    float xbuf[U];
    #pragma unroll
    for (int u = 0; u < U; ++u) xbuf[u] = xrow[u * HP];

    for (int t0 = 0; t0 < S_; t0 += U) {
        float xcur[U];
        #pragma unroll
        for (int u = 0; u < U; ++u) xcur[u] = xbuf[u];
        if (t0 + U < S_) {                        // uniform branch
            #pragma unroll
            for (int u = 0; u < U; ++u) xbuf[u] = xrow[(t0 + U + u) * HP];
        }
        #pragma unroll
        for (int u = 0; u < U; ++u) {
            float a = xcur[u];
            #pragma unroll
            for (int k = 0; k < H_; ++k) {
                float hk = __uint_as_float(
                    __builtin_amdgcn_readlane(__float_as_uint(h), k));
                a = fmaf(w[k], hk, a);
            }
            h = tanh_fast(a);
        }
    }

    // Final FC: out[o][b] = b_fc[o] + sum_k W_fc[o][k] * h[k];  out is (O,B,1)
    if (lane < O_) {
        float acc = bfc[lane];
        #pragma unroll
        for (int k = 0; k < H_; ++k) {
            float hk = __uint_as_float(
                __builtin_amdgcn_readlane(__float_as_uint(h), k));
            acc = fmaf(Wfc[lane * H_ + k], hk, acc);
        }
        out[lane * B_ + b] = acc;
    }
}

// ---------------------------------------------------------------------------
extern "C" void kernel_launch(void* const* d_in, const int* in_sizes, int n_in,
                              void* d_out, int out_size, void* d_ws, size_t ws_size,
                              hipStream_t stream)
{
    const float* x   = (const float*)d_in[0];
    const float* Wih = (const float*)d_in[1];
    const float* Whh = (const float*)d_in[2];
    const float* bih = (const float*)d_in[3];
    const float* bhh = (const float*)d_in[4];
    const float* Wfc = (const float*)d_in[5];
    const float* bfc = (const float*)d_in[6];
    float* out = (float*)d_out;
    float* xp  = (float*)d_ws;                    // needs B*S*32*4 = 16 MB

    k_inproj<<<dim3((B_ * S_) / 128), dim3(256), 0, stream>>>(x, Wih, bih, bhh, xp);
    k_scan  <<<dim3(B_),             dim3(32),  0, stream>>>(xp, Whh, Wfc, bfc, out);
}